// KernelMachine_40355512713599
// MI455X (gfx1250) — compile-verified
//
#include <hip/hip_runtime.h>

typedef float v2f __attribute__((ext_vector_type(2)));
typedef float v8f __attribute__((ext_vector_type(8)));

#define DIM    64
#define NSAMP  4096
#define MDIM   8
#define BATCH  8192
#define BT     64            // batch rows per block (4 row-groups x 16)
#define NT     16            // n-tile width (WMMA N)
#define NSPLIT 2             // n-reduction split across blocks (grid.y)
#define NRANGE (NSAMP / NSPLIT)  // 2048 samples per block
#define NQ     (NRANGE / 2)      // 1024 samples per wave-half
#define NTILES (NQ / NT)         // 64 tiles per wave-half

__global__ void km_zero_out(float* __restrict__ out, int n) {
    const int i = blockIdx.x * blockDim.x + threadIdx.x;
    if (i < n) out[i] = 0.0f;
}

// out[b,m] += sum_{n in block's range} scale*cos(x[b,:].Wf[n,:] + bf[n]) * Wp[n,m]
__global__ __launch_bounds__(256) void km_fused_kernel(
    const float* __restrict__ x,    // [BATCH, DIM]
    const float* __restrict__ Wf,   // [NSAMP, DIM]
    const float* __restrict__ bf,   // [NSAMP]
    const float* __restrict__ Wp,   // [NSAMP, MDIM]
    float* __restrict__ out)        // [BATCH, MDIM] (pre-zeroed, atomically accumulated)
{
    // [stage parity][n-half][16 x 64 tile] : 16 KB double-buffered staging
    __shared__ float sWf[2][2][NT * DIM];
    __shared__ float sOut[BT * MDIM];    // block-level partial accumulator

    const int tid  = threadIdx.x;
    const int lane = tid & 31;
    const int wave = tid >> 5;       // 0..7
    const int rg   = wave & 3;       // row-group within block (16 rows each)
    const int half = wave >> 2;      // which n-half of this block's range
    const int h    = lane >> 4;      // 0: K=0..1 / rows 0..7 ; 1: K=2..3 / rows 8..15
    const int l16  = lane & 15;

    const int bbase  = blockIdx.x * BT;
    const int nsbase = blockIdx.y * NRANGE;   // this block's n-range start

    // zero the block accumulator
    for (int i = tid; i < BT * MDIM; i += 256) sOut[i] = 0.0f;

    // ---- preload this lane's x-row as 16 A-fragments (K-step layout) ----
    // A (16x4 f32): lane l16 holds row l16; VGPR pair holds K = 4*kk + 2*h + {0,1}
    const int arow = bbase + rg * 16 + l16;
    const float* xr = x + arow * DIM + 2 * h;
    v2f a[16];
#pragma unroll
    for (int kk = 0; kk < 16; ++kk)
        a[kk] = *(const v2f*)(xr + 4 * kk);

    // per-lane partial output: 8 rows (this lane's C rows) x 8 match dims
    float acc[8][8];
#pragma unroll
    for (int r = 0; r < 8; ++r)
#pragma unroll
        for (int m = 0; m < 8; ++m) acc[r][m] = 0.0f;

    const float scale  = 0.022097086912079608f;  // sqrt(2/4096)
    const int   nstart = nsbase + half * NQ;

    // cooperative staging of one 16x64 Wf tile per n-half into LDS.
    // threads 0..127 stage the half-0 tile, 128..255 the half-1 tile.
    auto stage = [&](int tt) {
        const int g   = tid & 127;
        const int buf = tid >> 7;
        const int nb2 = nsbase + buf * NQ + tt * NT;
        const int idx = g * 8;            // 8 floats per thread
        const int rr  = idx >> 6;         // row within tile
        const int cc  = idx & 63;         // col within row
        const float4* src = (const float4*)(Wf + (nb2 + rr) * DIM + cc);
        float4* dst = (float4*)(&sWf[tt & 1][buf][idx]);
        dst[0] = src[0];
        dst[1] = src[1];
        if (tt + 1 < NTILES)   // WGP-scope prefetch of the tile after this one
            __builtin_prefetch(Wf + (nb2 + NT + rr) * DIM + cc, 0, 3);
    };

    stage(0);

    for (int t = 0; t < NTILES; ++t) {
        __syncthreads();                   // tile t visible; buf (t+1)&1 free
        if (t + 1 < NTILES) stage(t + 1);  // overlap staging with compute

        const int   nb   = nstart + t * NT;
        const int   ncol = nb + l16;       // this lane's column of the f tile
        const float bias = bf[ncol];
        const float4 wp0 = *(const float4*)(Wp + ncol * MDIM);
        const float4 wp1 = *(const float4*)(Wp + ncol * MDIM + 4);

        // ---- B fragments from LDS (pipelined ds_load_2addr_b64) ----
        // B (4x16 f32): lane l16 holds column l16 = Wf row (nb+l16); K = 4*kk+2*h+{0,1}
        const float* bsrc = &sWf[t & 1][half][l16 * DIM + 2 * h];
        v2f bfrag[16];
#pragma unroll
        for (int kk = 0; kk < 16; ++kk)
            bfrag[kk] = *(const v2f*)(bsrc + 4 * kk);

        // ---- GEMM1 tile: two independent WMMA chains over K ----
        v8f f0 = {}, f1 = {};
#pragma unroll
        for (int kk = 0; kk < 16; kk += 2) {
            f0 = __builtin_amdgcn_wmma_f32_16x16x4_f32(
                false, a[kk],     false, bfrag[kk],     (short)0, f0, false, false);
            f1 = __builtin_amdgcn_wmma_f32_16x16x4_f32(
                false, a[kk + 1], false, bfrag[kk + 1], (short)0, f1, false, false);
        }

        // ---- fused epilogue: merge chains + bias, cos, scale, fold g @ Wp ----
#pragma unroll
        for (int r = 0; r < 8; ++r) {
            const float gv = scale * __cosf(f0[r] + f1[r] + bias);
            acc[r][0] = fmaf(gv, wp0.x, acc[r][0]);
            acc[r][1] = fmaf(gv, wp0.y, acc[r][1]);
            acc[r][2] = fmaf(gv, wp0.z, acc[r][2]);
            acc[r][3] = fmaf(gv, wp0.w, acc[r][3]);
            acc[r][4] = fmaf(gv, wp1.x, acc[r][4]);
            acc[r][5] = fmaf(gv, wp1.y, acc[r][5]);
            acc[r][6] = fmaf(gv, wp1.z, acc[r][6]);
            acc[r][7] = fmaf(gv, wp1.w, acc[r][7]);
        }
    }

    // ---- reduce 16 lanes per row-half + 2 n-halves via native ds_add_f32 ----
#pragma unroll
    for (int r = 0; r < 8; ++r) {
        const int orow = rg * 16 + h * 8 + r;   // C-layout row of this lane
#pragma unroll
        for (int m = 0; m < 8; ++m)
            unsafeAtomicAdd(&sOut[orow * MDIM + m], acc[r][m]);
    }
    __syncthreads();

    // cross-block n-split reduction: native global_atomic_add_f32
    for (int i = tid; i < BT * MDIM; i += 256)
        unsafeAtomicAdd(&out[bbase * MDIM + i], sOut[i]);
}

extern "C" void kernel_launch(void* const* d_in, const int* in_sizes, int n_in,
                              void* d_out, int out_size, void* d_ws, size_t ws_size,
                              hipStream_t stream) {
    (void)in_sizes; (void)n_in; (void)d_ws; (void)ws_size;
    const float* x  = (const float*)d_in[0];
    const float* Wf = (const float*)d_in[1];
    const float* bf = (const float*)d_in[2];
    const float* Wp = (const float*)d_in[3];
    float* out = (float*)d_out;

    const int n_out = BATCH * MDIM;   // 65536
    km_zero_out<<<(n_out + 255) / 256, 256, 0, stream>>>(out, n_out);

    dim3 grid(BATCH / BT, NSPLIT);    // 128 x 2 = 256 blocks
    dim3 block(256);                  // 8 wave32
    km_fused_kernel<<<grid, block, 0, stream>>>(x, Wf, bf, Wp, out);
}